// GroupAttention_9509057593557
// MI455X (gfx1250) — compile-verified
//
#include <hip/hip_runtime.h>
#include <math.h>

#define B_ 4
#define S_ 2048
#define D_ 256

typedef __attribute__((ext_vector_type(2))) float v2f;
typedef __attribute__((ext_vector_type(8))) float v8f;

// ---------------- 1) LayerNorm (unbiased std, eps on std) ----------------
__global__ void ga_ln_kernel(const float* __restrict__ ctx,
                             const float* __restrict__ ln_a,
                             const float* __restrict__ ln_b,
                             float* __restrict__ xout) {
    __shared__ float red[256];
    int row = blockIdx.x;            // b*S + s
    int t = threadIdx.x;             // column 0..255
    float v = ctx[(size_t)row * D_ + t];
    red[t] = v; __syncthreads();
    for (int off = 128; off > 0; off >>= 1) {
        if (t < off) red[t] += red[t + off];
        __syncthreads();
    }
    float mean = red[0] * (1.0f / (float)D_);
    __syncthreads();
    float d = v - mean;
    red[t] = d * d; __syncthreads();
    for (int off = 128; off > 0; off >>= 1) {
        if (t < off) red[t] += red[t + off];
        __syncthreads();
    }
    float var = red[0] * (1.0f / (float)(D_ - 1));  // ddof=1
    float stdv = sqrtf(var);
    xout[(size_t)row * D_ + t] = ln_a[t] * d / (stdv + 1e-6f) + ln_b[t];
}

// ---------------- 2) Y = X @ W^T + bias via V_WMMA_F32_16X16X4_F32 ----------
// One wave computes one 16x16 output tile; 64 WMMA issues (K=256 in steps of 4).
// A-frag: lane<16 holds (M=lane, K={0,1}); lane>=16 holds (M=lane-16, K={2,3}).
// B-frag (= W^T tile): same lane pattern with N=lane&15.
// C/D: VGPR r -> M = r + (lane>=16 ? 8 : 0), N = lane&15.
__global__ void ga_gemm_wmma(const float* __restrict__ X,
                             const float* __restrict__ W,
                             const float* __restrict__ bias,
                             float* __restrict__ Y,
                             int nrows) {
    int wave = threadIdx.x >> 5;
    int lane = threadIdx.x & 31;
    int tile = blockIdx.x * (blockDim.x >> 5) + wave;
    int total = (nrows >> 4) * (D_ >> 4);
    if (tile >= total) return;
    int tn = tile & ((D_ >> 4) - 1);
    int tm = tile >> 4;
    int half = lane >> 4;
    int mr   = lane & 15;
    const float* xrow = X + (size_t)(tm * 16 + mr) * D_ + 2 * half;
    const float* wrow = W + (size_t)(tn * 16 + mr) * D_ + 2 * half;
    v8f acc = {};
    for (int kk = 0; kk < D_ / 4; ++kk) {
        v2f a = *(const v2f*)(xrow + kk * 4);
        v2f b = *(const v2f*)(wrow + kk * 4);
        acc = __builtin_amdgcn_wmma_f32_16x16x4_f32(false, a, false, b,
                                                    (short)0, acc, false, false);
    }
    float bv  = bias[tn * 16 + mr];
    int rbase = tm * 16 + half * 8;
    int col   = tn * 16 + mr;
    for (int r = 0; r < 8; ++r)
        Y[(size_t)(rbase + r) * D_ + col] = acc[r] + bv;
}

// ---------------- 3) adjacent scores: s_up[m]=q_m.k_{m+1}/256, s_dn[m]=q_{m+1}.k_m/256
__global__ void ga_adj_scores(const float* __restrict__ Q,
                              const float* __restrict__ K,
                              float* __restrict__ s_up,
                              float* __restrict__ s_dn) {
    int wave = threadIdx.x >> 5;
    int lane = threadIdx.x & 31;
    int w = blockIdx.x * (blockDim.x >> 5) + wave;
    int total = B_ * (S_ - 1);
    if (w >= total) return;
    int b = w / (S_ - 1);
    int m = w - b * (S_ - 1);
    const float* qm  = Q + (size_t)(b * S_ + m) * D_;
    const float* qm1 = Q + (size_t)(b * S_ + m + 1) * D_;
    const float* km  = K + (size_t)(b * S_ + m) * D_;
    const float* km1 = K + (size_t)(b * S_ + m + 1) * D_;
    float su = 0.f, sd = 0.f;
    int e0 = lane * 8;
    for (int e = e0; e < e0 + 8; e += 4) {
        float4 a  = *(const float4*)(qm + e);
        float4 bb = *(const float4*)(km1 + e);
        su += a.x * bb.x + a.y * bb.y + a.z * bb.z + a.w * bb.w;
        float4 c  = *(const float4*)(qm1 + e);
        float4 dd = *(const float4*)(km + e);
        sd += c.x * dd.x + c.y * dd.y + c.z * dd.z + c.w * dd.w;
    }
    for (int off = 16; off > 0; off >>= 1) {
        su += __shfl_xor(su, off, 32);
        sd += __shfl_xor(sd, off, 32);
    }
    if (lane == 0) {
        s_up[b * S_ + m] = su * (1.0f / 256.0f);
        s_dn[b * S_ + m] = sd * (1.0f / 256.0f);
    }
}

// ---------------- 4) per-row 2-way softmax on the masked diagonals -----------
__global__ void ga_adj_softmax(const float* __restrict__ s_up,
                               const float* __restrict__ s_dn,
                               const int* __restrict__ eos,
                               const float* __restrict__ prior,
                               float* __restrict__ soft_l,
                               float* __restrict__ soft_r,
                               float* __restrict__ diag) {
    int idx = blockIdx.x * blockDim.x + threadIdx.x;
    if (idx >= B_ * S_) return;
    int b = idx >> 11;
    int q = idx & (S_ - 1);
    int mL = (q > 0)      ? eos[(size_t)(b * S_ + q) * S_ + (q - 1)] : 0;
    int mR = (q < S_ - 1) ? eos[(size_t)(b * S_ + q) * S_ + (q + 1)] : 0;
    float sL = (q > 0)      ? s_dn[b * S_ + q - 1] : 0.f;
    float sR = (q < S_ - 1) ? s_up[b * S_ + q]     : 0.f;
    float softL, softR, dd;
    if (mL && mR) {
        float mx = fmaxf(sL, sR);
        float eL = expf(sL - mx), eR = expf(sR - mx);
        float inv = 1.f / (eL + eR);
        softL = eL * inv; softR = eR * inv; dd = 0.f;
    } else if (mL) { softL = 1.f; softR = 0.f; dd = 0.f; }
    else if (mR)   { softL = 0.f; softR = 1.f; dd = 0.f; }
    else { softL = softR = dd = 1.0f / (float)S_; }  // fully-masked row: uniform
    soft_l[idx] = softL;
    soft_r[idx] = softR;
    float pr = prior[(size_t)(b * S_ + q) * S_ + q];
    diag[idx] = pr + (1.f - pr) * sqrtf(dd * dd + 1e-9f);
}

// ---------------- 5) superdiagonal log terms L_m --------------------------
__global__ void ga_sup_log(const float* __restrict__ soft_l,
                           const float* __restrict__ soft_r,
                           const float* __restrict__ prior,
                           float* __restrict__ Lbuf) {
    int idx = blockIdx.x * blockDim.x + threadIdx.x;
    if (idx >= B_ * S_) return;
    int b = idx >> 11;
    int m = idx & (S_ - 1);
    float Lv = 0.f;
    if (m < S_ - 1) {
        float v  = sqrtf(soft_r[idx] * soft_l[idx + 1] + 1e-9f);
        float pr = prior[(size_t)(b * S_ + m) * S_ + (m + 1)];
        float nb = pr + (1.f - pr) * v;
        Lv = logf(nb + 1e-9f);
    }
    Lbuf[idx] = Lv;
}

// ---------------- 6) exclusive prefix sum C[i] = sum_{m<i} L_m, per batch ----
__global__ void ga_scan(const float* __restrict__ Lbuf, float* __restrict__ C) {
    __shared__ float ssum[256];
    int b = blockIdx.x;
    int t = threadIdx.x;
    int base = t * 8;
    float v[8];
    float tot = 0.f;
    for (int e = 0; e < 8; ++e) {
        v[e] = tot;                                   // chunk-local exclusive
        tot += Lbuf[b * S_ + base + e];
    }
    ssum[t] = tot; __syncthreads();
    for (int off = 1; off < 256; off <<= 1) {         // Hillis-Steele inclusive
        float x = (t >= off) ? ssum[t - off] : 0.f;
        __syncthreads();
        ssum[t] += x;
        __syncthreads();
    }
    float cb = (t > 0) ? ssum[t - 1] : 0.f;
    for (int e = 0; e < 8; ++e)
        C[b * S_ + base + e] = cb + v[e];
}

// ---------------- 7) output sweep (bandwidth bound): 67 MB of fp32 ----------
__global__ void ga_out(const float* __restrict__ C,
                       const float* __restrict__ diag,
                       float* __restrict__ out) {
    int row = blockIdx.y;          // b*S + i
    int b = row >> 11;
    int i = row & (S_ - 1);
    int j0 = (blockIdx.x * blockDim.x + threadIdx.x) * 4;
    float ci = C[row];
    float4 cj4 = *(const float4*)(C + b * S_ + j0);
    float cj[4] = {cj4.x, cj4.y, cj4.z, cj4.w};
    float4 o;
    float* op = &o.x;
    for (int e = 0; e < 4; ++e) {
        int j = j0 + e;
        if (j == i) {
            op[e] = diag[row];
        } else {
            float ds = (j > i) ? (cj[e] - ci) : (ci - cj[e]);
            op[e] = expf(ds) + 1e-9f;
        }
    }
    *(float4*)(out + (size_t)row * S_ + j0) = o;
}

extern "C" void kernel_launch(void* const* d_in, const int* in_sizes, int n_in,
                              void* d_out, int out_size, void* d_ws, size_t ws_size,
                              hipStream_t stream) {
    const float* ctx   = (const float*)d_in[0];
    const float* prior = (const float*)d_in[1];
    const float* Wq    = (const float*)d_in[2];
    const float* bq    = (const float*)d_in[3];
    const float* Wk    = (const float*)d_in[4];
    const float* bk    = (const float*)d_in[5];
    const float* ln_a  = (const float*)d_in[6];
    const float* ln_b  = (const float*)d_in[7];
    const int*   eos   = (const int*)d_in[8];
    float* out = (float*)d_out;

    float* ws = (float*)d_ws;
    const size_t NTOK = (size_t)B_ * S_;          // 8192
    const size_t NXD  = NTOK * D_;                // 2,097,152
    float* X      = ws;
    float* Q      = X + NXD;
    float* K      = Q + NXD;
    float* s_up   = K + NXD;
    float* s_dn   = s_up   + NTOK;
    float* soft_l = s_dn   + NTOK;
    float* soft_r = soft_l + NTOK;
    float* diag   = soft_r + NTOK;
    float* Lbuf   = diag   + NTOK;
    float* C      = Lbuf   + NTOK;

    // 1) LayerNorm: one 256-thread block (8 waves) per row
    ga_ln_kernel<<<(int)NTOK, 256, 0, stream>>>(ctx, ln_a, ln_b, X);

    // 2) Q/K projections with fp32 WMMA: 8192 tiles, 8 waves per block
    ga_gemm_wmma<<<1024, 256, 0, stream>>>(X, Wq, bq, Q, (int)NTOK);
    ga_gemm_wmma<<<1024, 256, 0, stream>>>(X, Wk, bk, K, (int)NTOK);

    // 3) adjacent-diagonal dot products: one wave per (b, m)
    ga_adj_scores<<<1024, 256, 0, stream>>>(Q, K, s_up, s_dn);

    // 4) masked 2-way softmax per row + diag of neibor
    ga_adj_softmax<<<(int)(NTOK / 256), 256, 0, stream>>>(s_up, s_dn, eos, prior,
                                                          soft_l, soft_r, diag);

    // 5) superdiagonal log terms
    ga_sup_log<<<(int)(NTOK / 256), 256, 0, stream>>>(soft_l, soft_r, prior, Lbuf);

    // 6) per-batch exclusive scan
    ga_scan<<<B_, 256, 0, stream>>>(Lbuf, C);

    // 7) output: exp(C[max]-C[min]) + 1e-9 off-diagonal, neibor diag on-diagonal
    ga_out<<<dim3(S_ / (128 * 4), (int)NTOK), 128, 0, stream>>>(C, diag, out);

    (void)in_sizes; (void)n_in; (void)out_size; (void)ws_size;
}